// RPN_48507360641431
// MI455X (gfx1250) — compile-verified
//
#include <hip/hip_runtime.h>
#include <hip/hip_bf16.h>

typedef __attribute__((ext_vector_type(2))) float v2f;
typedef __attribute__((ext_vector_type(8))) float v8f;

#define FSZ      50
#define NPOS     2500        // 50*50 positions (batch 7 only)
#define CIN      512
#define COUT     512
#define KDIM     4608        // 512 * 9
#define NANCH    22500       // 2500 * 9
#define SORT_N   32768
#define PRE_NMS  6000
#define POST_NMS 300

#define KCHUNK   32
#define BSROW    36          // padded row stride (floats): conflict-free b64 reads

// ---------------------------------------------------------------------------
// Kernel 1: 3x3 conv as implicit-im2col GEMM on f32 WMMA (16x16x4), fused
// bias + leaky-ReLU epilogue. Grid: x = N-tiles of 64 (40), y = M-tiles of 128
// (4). Block = 256 threads = 8 waves; wave w owns M rows [..+16); the n-major
// 64x32 f32 B-tile in LDS is shared by all 8 waves (B depends only on N,K).
// n-major layout => each lane's (B[k][n], B[k+1][n]) fragment is one aligned
// ds_load_b64 directly into the even VGPR pair WMMA requires (no repacking).
// ---------------------------------------------------------------------------
__global__ __launch_bounds__(256) void conv3x3_wmma(
    const float* __restrict__ in,    // (8,512,50,50) -> use b=7
    const float* __restrict__ W,     // (512,4608) = OIHW flat
    const float* __restrict__ bias,  // (512)
    float* __restrict__ feat)        // (512,2500)
{
  __shared__ float Bs[64][BSROW];    // [n][k], k-pad 32->36

  const int tid  = threadIdx.x;
  const int wave = tid >> 5;
  const int lane = tid & 31;
  const int l16  = lane & 15;
  const int lhi  = lane >> 4;          // 0 or 1 (half-wave)
  const int koff = lhi * 2;            // half-wave K offset within group of 4

  const int nbase = blockIdx.x * 64;
  const int mbase = blockIdx.y * 128 + wave * 16;

  const float* in7  = in + (size_t)7 * CIN * NPOS;
  const int    arow = mbase + l16;                    // < 512 always
  const float* Wrow = W + (size_t)arow * KDIM;

  v8f acc[4];
#pragma unroll
  for (int q = 0; q < 4; ++q)
#pragma unroll
    for (int e = 0; e < 8; ++e) acc[q][e] = 0.f;

  for (int K = 0; K < KDIM; K += KCHUNK) {
    __syncthreads();
    // cooperative im2col load of B tile: Bs[n][k] = in7[ci, y+ky-1, x+kx-1]
    // consecutive threads -> consecutive n => coalesced global reads;
    // LDS store stride BSROW floats.
#pragma unroll
    for (int e = tid; e < KCHUNK * 64; e += 256) {
      const int kk = e >> 6, nn = e & 63;
      const int k  = K + kk;
      const int ci = k / 9;
      const int r  = k - ci * 9;
      const int ky = r / 3;
      const int kx = r - ky * 3;
      const int p  = nbase + nn;
      float v = 0.f;
      if (p < NPOS) {
        const int y = p / FSZ, x = p - y * FSZ;
        const int yy = y + ky - 1, xx = x + kx - 1;
        if ((unsigned)yy < (unsigned)FSZ && (unsigned)xx < (unsigned)FSZ)
          v = in7[(ci * FSZ + yy) * FSZ + xx];
      }
      Bs[nn][kk] = v;
    }
    __syncthreads();

#pragma unroll
    for (int kk = 0; kk < KCHUNK; kk += 4) {
      // A fragment: lane (l16,lhi) -> row mbase+l16, K pair {K+kk+koff, +1}
      const float2 av = *(const float2*)(Wrow + K + kk + koff);
      v2f a; a.x = av.x; a.y = av.y;
#pragma unroll
      for (int q = 0; q < 4; ++q) {
        const float2 bv = *(const float2*)(&Bs[q * 16 + l16][kk + koff]);
        v2f b; b.x = bv.x; b.y = bv.y;
        acc[q] = __builtin_amdgcn_wmma_f32_16x16x4_f32(
            false, a, false, b, (short)0, acc[q], false, false);
      }
    }
  }

  // epilogue: D layout — lanes 0-15: VGPR v -> M=v, lanes 16-31: M=v+8
#pragma unroll
  for (int q = 0; q < 4; ++q) {
#pragma unroll
    for (int v = 0; v < 8; ++v) {
      const int co = mbase + v + lhi * 8;
      const int p  = nbase + q * 16 + l16;
      if (p < NPOS) {
        float y = acc[q][v] + bias[co];
        feat[(size_t)co * NPOS + p] = (y >= 0.f) ? y : 0.01f * y;
      }
    }
  }
}

// ---------------------------------------------------------------------------
// Kernel 2: init sort keys to +max (padding sorts last)
// ---------------------------------------------------------------------------
__global__ void init_keys(unsigned long long* __restrict__ keys) {
  int i = blockIdx.x * blockDim.x + threadIdx.x;
  if (i < SORT_N) keys[i] = ~0ull;
}

// ---------------------------------------------------------------------------
// Kernel 3: 1x1 heads (reg 36ch, cls 18ch) + softmax score + box decode.
// One thread per spatial position p; weights staged in LDS in chunks.
// Emits boxes[22500][4] and packed sort keys.
// ---------------------------------------------------------------------------
#define CCHUNK 128
__global__ __launch_bounds__(256) void head_decode(
    const float* __restrict__ feat,   // (512,2500)
    const float* __restrict__ reg_w,  // (36,512)
    const float* __restrict__ reg_b,  // (36)
    const float* __restrict__ cls_w,  // (18,512)
    const float* __restrict__ cls_b,  // (18)
    float* __restrict__ boxes,        // (22500,4)
    unsigned long long* __restrict__ keys)
{
  __shared__ float wreg[36][CCHUNK];
  __shared__ float wcls[18][CCHUNK];

  const int p = blockIdx.x * 256 + threadIdx.x;

  float racc[36], cacc[18];
#pragma unroll
  for (int o = 0; o < 36; ++o) racc[o] = reg_b[o];
#pragma unroll
  for (int o = 0; o < 18; ++o) cacc[o] = cls_b[o];

  for (int c0 = 0; c0 < CIN; c0 += CCHUNK) {
    __syncthreads();
    for (int e = threadIdx.x; e < 36 * CCHUNK; e += 256)
      wreg[e >> 7][e & 127] = reg_w[(e >> 7) * CIN + c0 + (e & 127)];
    for (int e = threadIdx.x; e < 18 * CCHUNK; e += 256)
      wcls[e >> 7][e & 127] = cls_w[(e >> 7) * CIN + c0 + (e & 127)];
    __syncthreads();
    if (p < NPOS) {
      for (int c = 0; c < CCHUNK; ++c) {
        const float f = feat[(size_t)(c0 + c) * NPOS + p];
#pragma unroll
        for (int o = 0; o < 36; ++o) racc[o] = fmaf(f, wreg[o][c], racc[o]);
#pragma unroll
        for (int o = 0; o < 18; ++o) cacc[o] = fmaf(f, wcls[o][c], cacc[o]);
      }
    }
  }

  if (p >= NPOS) return;

  // anchors (reference quirk preserved: cx indexed by p/50, cy by p%50)
  const int iy = p / FSZ, ix = p - iy * FSZ;
  const float cx = 16.f * iy + 8.f;
  const float cy = 16.f * ix + 8.f;
  const float rsv[3] = {0.5f, 1.f, 2.f};
  const float ssv[3] = {8.f, 16.f, 32.f};

#pragma unroll
  for (int a = 0; a < 9; ++a) {
    const int r = a / 3, s = a - r * 3;
    const float sq = sqrtf(rsv[r]);
    const float ha = 16.f * ssv[s] * sq;
    const float wa = 16.f * ssv[s] / sq;
    const float x1a = cx - 0.5f * wa;
    const float y1a = cy - 0.5f * ha;

    const float p0 = racc[a * 4 + 0], p1 = racc[a * 4 + 1];
    const float p2 = racc[a * 4 + 2], p3 = racc[a * 4 + 3];
    const float c0v = cacc[a * 2 + 0], c1v = cacc[a * 2 + 1];
    const float score = 1.f / (1.f + expf(c0v - c1v));  // softmax[:,1]

    const float rx1 = fminf(fmaxf(p0 + x1a, 0.f), 799.f);
    const float ry1 = fminf(fmaxf(p1 + y1a, 0.f), 799.f);
    const float rx2 = fminf(fmaxf(p0 + x1a + p2 + wa, 0.f), 799.f);
    const float ry2 = fminf(fmaxf(p1 + y1a + p3 + ha, 0.f), 799.f);
    const float w = p2 + wa, h = p3 + ha;
    const bool valid = (w >= 16.f) && (h >= 16.f);
    const float sc = valid ? score : -__builtin_inff();

    const int n = p * 9 + a;
    float4 bb; bb.x = rx1; bb.y = ry1; bb.z = rx2; bb.w = ry2;
    ((float4*)boxes)[n] = bb;

    // key: ascending u64 sort == descending score, ties -> ascending index
    const unsigned bits = __float_as_uint(sc);
    const unsigned ord  = (bits & 0x80000000u) ? ~bits : (bits | 0x80000000u);
    keys[n] = (((unsigned long long)(~ord)) << 32) | (unsigned)n;
  }
}

// ---------------------------------------------------------------------------
// Kernel 4: bitonic sort step (ascending), 32768 u64 keys
// ---------------------------------------------------------------------------
__global__ void bitonic_step(unsigned long long* __restrict__ keys, int k, int j) {
  const int i = blockIdx.x * blockDim.x + threadIdx.x;
  const int ixj = i ^ j;
  if (ixj > i && i < SORT_N) {
    unsigned long long a = keys[i], b = keys[ixj];
    const bool up = ((i & k) == 0);
    if (up ? (a > b) : (a < b)) { keys[i] = b; keys[ixj] = a; }
  }
}

// ---------------------------------------------------------------------------
// Kernel 5: gather top-6000 boxes in score order
// ---------------------------------------------------------------------------
__global__ void gather_boxes(const unsigned long long* __restrict__ keys,
                             const float* __restrict__ boxes,
                             float* __restrict__ nmsbox) {
  const int r = blockIdx.x * blockDim.x + threadIdx.x;
  if (r < PRE_NMS) {
    const unsigned n = (unsigned)(keys[r] & 0xFFFFFFFFull);
    float4 b; b.x = 0.f; b.y = 0.f; b.z = 0.f; b.w = 0.f;
    if (n < (unsigned)NANCH) b = ((const float4*)boxes)[n];
    ((float4*)nmsbox)[r] = b;
  }
}

// ---------------------------------------------------------------------------
// Kernel 6: NMS (replicates reference exactly, incl. yy2 = MAX quirk) +
// argsort(where(keep, arange, P))[:300] selection + final [x,y,w,h] output.
// Single block of 1024 threads; keep flags in LDS.
// ---------------------------------------------------------------------------
__global__ __launch_bounds__(1024) void nms_out(
    const float* __restrict__ bx,  // (6000,4) x1,y1,x2,y2 in score order
    float* __restrict__ out)       // (300,4)
{
  __shared__ unsigned char keep[PRE_NMS];
  __shared__ int sel[POST_NMS];

  const int tid = threadIdx.x;

  float x1[6], y1[6], x2[6], y2[6], ar[6];
#pragma unroll
  for (int t = 0; t < 6; ++t) {
    const int j = tid + t * 1024;
    if (j < PRE_NMS) {
      const float4 b = ((const float4*)bx)[j];
      x1[t] = b.x; y1[t] = b.y; x2[t] = b.z; y2[t] = b.w;
      ar[t] = (b.z - b.x + 1.f) * (b.w - b.y + 1.f);
    } else {
      x1[t] = y1[t] = x2[t] = y2[t] = 0.f; ar[t] = 1.f;
    }
  }
  for (int i = tid; i < PRE_NMS; i += 1024) keep[i] = 1;
  __syncthreads();

  for (int i = 0; i < PRE_NMS; ++i) {
    if (keep[i]) {                       // uniform across block
      const float4 b = ((const float4*)bx)[i];
      const float ai = (b.z - b.x + 1.f) * (b.w - b.y + 1.f);
#pragma unroll
      for (int t = 0; t < 6; ++t) {
        const int j = tid + t * 1024;
        if (j < PRE_NMS && j > i && keep[j]) {
          const float xx1 = fmaxf(b.x, x1[t]);
          const float yy1 = fmaxf(b.y, y1[t]);
          const float xx2 = fminf(b.z, x2[t]);
          const float yy2 = fmaxf(b.w, y2[t]);   // reference uses maximum here
          const float w = fmaxf(0.f, xx2 - xx1 + 1.f);
          const float h = fmaxf(0.f, yy2 - yy1 + 1.f);
          const float inter = w * h;
          const float ov = inter / (ai + ar[t] - inter);
          if (ov > 0.7f) keep[j] = 0;
        }
      }
    }
    __syncthreads();
  }

  if (tid == 0) {
    int cnt = 0;
    for (int r = 0; r < PRE_NMS && cnt < POST_NMS; ++r)
      if (keep[r]) sel[cnt++] = r;
    for (int r = 0; r < PRE_NMS && cnt < POST_NMS; ++r)
      if (!keep[r]) sel[cnt++] = r;
  }
  __syncthreads();

  if (tid < POST_NMS) {
    const int r = sel[tid];
    const float4 b = ((const float4*)bx)[r];
    out[tid * 4 + 0] = b.x;
    out[tid * 4 + 1] = b.y;
    out[tid * 4 + 2] = b.z - b.x + 1.f;
    out[tid * 4 + 3] = b.w - b.y + 1.f;
  }
}

// ---------------------------------------------------------------------------
extern "C" void kernel_launch(void* const* d_in, const int* in_sizes, int n_in,
                              void* d_out, int out_size, void* d_ws, size_t ws_size,
                              hipStream_t stream) {
  const float* in_features = (const float*)d_in[0];
  const float* conv_w      = (const float*)d_in[1];
  const float* conv_b      = (const float*)d_in[2];
  const float* reg_w       = (const float*)d_in[3];
  const float* reg_b       = (const float*)d_in[4];
  const float* cls_w       = (const float*)d_in[5];
  const float* cls_b       = (const float*)d_in[6];
  float* out = (float*)d_out;

  char* ws = (char*)d_ws;
  float* feat  = (float*)(ws + 0);                    // 512*2500*4   = 5,120,000 B
  float* boxes = (float*)(ws + 5120000);              // 22500*4*4    =   360,000 B
  unsigned long long* keys = (unsigned long long*)(ws + 5480448);  // 32768*8 = 262,144 B
  float* nmsbox = (float*)(ws + 5742592);             // 6000*4*4     =    96,000 B

  // 1) conv3x3 + leaky relu via f32 WMMA (batch 7 only)
  conv3x3_wmma<<<dim3(40, 4), 256, 0, stream>>>(in_features, conv_w, conv_b, feat);

  // 2) key init (padding entries sort last)
  init_keys<<<SORT_N / 256, 256, 0, stream>>>(keys);

  // 3) 1x1 heads + decode + key pack
  head_decode<<<(NPOS + 255) / 256, 256, 0, stream>>>(
      feat, reg_w, reg_b, cls_w, cls_b, boxes, keys);

  // 4) full bitonic sort of 32768 u64 keys (descending score, stable index)
  for (int k = 2; k <= SORT_N; k <<= 1)
    for (int j = k >> 1; j > 0; j >>= 1)
      bitonic_step<<<SORT_N / 256, 256, 0, stream>>>(keys, k, j);

  // 5) gather top-6000 boxes
  gather_boxes<<<(PRE_NMS + 255) / 256, 256, 0, stream>>>(keys, boxes, nmsbox);

  // 6) NMS + selection + output
  nms_out<<<1, 1024, 0, stream>>>(nmsbox, out);
}